// ProviderGAT_91319594647925
// MI455X (gfx1250) — compile-verified
//
#include <hip/hip_runtime.h>
#include <math.h>

#define H 4
#define D 512
#define NCODES 100000
#define NSPEC 1024
#define SLOPE 0.2f
#define B1 512          // blocks for the two streaming passes
#define CHUNK 196       // ceil(100000/512)
#define BSTR 516        // LDS row stride for B matrix (bank-conflict free: 516 % 64 = 4)

typedef __attribute__((ext_vector_type(2))) float v2f;
typedef __attribute__((ext_vector_type(4))) float v4f;
typedef __attribute__((ext_vector_type(8))) float v8f;

#if defined(__gfx1250__) && __has_builtin(__builtin_amdgcn_wmma_f32_16x16x4_f32)
#define USE_WMMA 1
#else
#define USE_WMMA 0
#endif

// ---------------------------------------------------------------------------
// Kernel 0: v2[h,d] = W[h,d,:]·a2[h,:] ; t[h,d] = prov[d] * (W[h,d,:]·a1[h,:])
// (prov_score[h] = sum_d t[h,d], reduced redundantly per-block in score kernel)
// 64 blocks: (h, 32-row slab). Wave-per-row, coalesced b128 row reads,
// shfl-tree reduction, no LDS reduce.
// ---------------------------------------------------------------------------
__global__ void prep_kernel(const float* __restrict__ prov,
                            const float* __restrict__ W,
                            const float* __restrict__ a1,
                            const float* __restrict__ a2,
                            float* __restrict__ v2g,
                            float* __restrict__ tt) {
  __shared__ float a1h[D];
  __shared__ float a2h[D];
  const int h  = blockIdx.x >> 4;
  const int d0 = (blockIdx.x & 15) * 32;
  const int tid = threadIdx.x;
  a1h[tid]       = a1[h * D + tid];
  a1h[tid + 256] = a1[h * D + tid + 256];
  a2h[tid]       = a2[h * D + tid];
  a2h[tid + 256] = a2[h * D + tid + 256];
  __syncthreads();
  const int wave = tid >> 5, lane = tid & 31;
  for (int rl = wave; rl < 32; rl += 8) {
    const int d = d0 + rl;
    const float* base = W + (size_t)(h * D + d) * D;
    float p1 = 0.f, p2 = 0.f;
#pragma unroll
    for (int j = 0; j < 4; ++j) {
      const int k0 = j * 128 + lane * 4;
      v4f wv = *(const v4f*)(base + k0);
      p1 += wv.x * a1h[k0] + wv.y * a1h[k0 + 1] + wv.z * a1h[k0 + 2] + wv.w * a1h[k0 + 3];
      p2 += wv.x * a2h[k0] + wv.y * a2h[k0 + 1] + wv.z * a2h[k0 + 2] + wv.w * a2h[k0 + 3];
    }
#pragma unroll
    for (int off = 16; off; off >>= 1) {
      p1 += __shfl_xor(p1, off, 32);
      p2 += __shfl_xor(p2, off, 32);
    }
    if (lane == 0) {
      v2g[h * D + d] = p2;
      tt[h * D + d]  = prov[d] * p1;
    }
  }
}

// ---------------------------------------------------------------------------
// Kernel 1 (streaming pass 1): e[n,h] = leaky(c[h] + code[n]·v2[h]) + block max.
// Per wave: one 16-code tile via V_WMMA_F32_16X16X4_F32, K looped 128x.
//   A (16x4 f32): lane L holds row (L&15), K-half (L>>4): global b64 loads.
//   B (4x16 f32): v2 zero-padded to 16 cols in LDS (stride 516 -> no conflicts).
//   C (16x16 f32): cols 0..3 = per-head scores.
// ---------------------------------------------------------------------------
__global__ void score_kernel(const float* __restrict__ code,
                             const float* __restrict__ v2g,
                             const float* __restrict__ tt,
                             float* __restrict__ e4,
                             float* __restrict__ blkmax) {
  __shared__ float lds_b[16 * BSTR];
  __shared__ float cpart[256];
  __shared__ float csh[H];
  __shared__ float wmaxsh[8 * H];
  const int tid = threadIdx.x;
  const int wave = tid >> 5, lane = tid & 31;

  // provider score c[h] = sum_d tt[h,d] (cheap redundant per-block reduce)
  {
    const int h = tid >> 6, i = tid & 63;
    float p = 0.f;
    for (int d = i; d < D; d += 64) p += tt[h * D + d];
    cpart[tid] = p;
  }
  // B matrix into LDS: rows 0..3 = v2, rows 4..15 = 0
  for (int idx = tid; idx < 16 * BSTR; idx += 256) {
    const int col = idx / BSTR, kk = idx - col * BSTR;
    lds_b[idx] = (col < H && kk < D) ? v2g[col * D + kk] : 0.f;
  }
  __syncthreads();
  if (tid < H) {
    float c = 0.f;
    for (int j = 0; j < 64; ++j) c += cpart[tid * 64 + j];
    csh[tid] = c;
  }
  __syncthreads();

  const int n0b = blockIdx.x * CHUNK;
  const int n1b = min(n0b + CHUNK, NCODES);
  const int ll = lane & 15, hi = lane >> 4;

#if USE_WMMA
  float mlocal = -INFINITY;
  for (int t = wave; t * 16 < CHUNK; t += 8) {
    const int n0 = n0b + t * 16;
    if (n0 >= n1b) break;
    int row = n0 + ll;
    if (row > NCODES - 1) row = NCODES - 1;     // clamp tail; masked below
    const float* aptr = code + (size_t)row * D + hi * 2;
    const float* bptr = lds_b + ll * BSTR + hi * 2;
    v8f acc = {0.f, 0.f, 0.f, 0.f, 0.f, 0.f, 0.f, 0.f};
#pragma unroll 4
    for (int k = 0; k < D; k += 4) {
      v2f a = *(const v2f*)(aptr + k);
      v2f b = *(const v2f*)(bptr + k);
      acc = __builtin_amdgcn_wmma_f32_16x16x4_f32(false, a, false, b,
                                                  (short)0, acc, false, false);
    }
    // C layout: lane j holds column N=j; VGPR r = row r (lanes<16) / r+8 (lanes>=16)
    if (ll < H) {
      const int hh = ll;
      const float cH = csh[hh];
      const int mbase = hi * 8;
#pragma unroll
      for (int r = 0; r < 8; ++r) {
        const int n = n0 + mbase + r;
        if (n < n1b) {
          const float s = acc[r] + cH;
          const float e = (s > 0.f) ? s : SLOPE * s;
          e4[(size_t)n * H + hh] = e;
          mlocal = fmaxf(mlocal, e);
        }
      }
    }
  }
  mlocal = fmaxf(mlocal, __shfl_xor(mlocal, 16, 32));
  if (lane < H) wmaxsh[wave * H + lane] = mlocal;
#else
  float mh0 = -INFINITY, mh1 = -INFINITY, mh2 = -INFINITY, mh3 = -INFINITY;
  for (int t = wave; t * 16 < CHUNK; t += 8) {
    const int n0 = n0b + t * 16;
    if (n0 >= n1b) break;
    int row = n0 + ll;
    if (row > NCODES - 1) row = NCODES - 1;
    const float* aptr = code + (size_t)row * D + hi * 2;
    float s0 = 0, s1 = 0, s2 = 0, s3 = 0;
    for (int k = 0; k < D; k += 4) {
      v2f a = *(const v2f*)(aptr + k);
      const int kk = k + hi * 2;
      s0 += a.x * lds_b[0 * BSTR + kk] + a.y * lds_b[0 * BSTR + kk + 1];
      s1 += a.x * lds_b[1 * BSTR + kk] + a.y * lds_b[1 * BSTR + kk + 1];
      s2 += a.x * lds_b[2 * BSTR + kk] + a.y * lds_b[2 * BSTR + kk + 1];
      s3 += a.x * lds_b[3 * BSTR + kk] + a.y * lds_b[3 * BSTR + kk + 1];
    }
    s0 += __shfl_xor(s0, 16, 32); s1 += __shfl_xor(s1, 16, 32);
    s2 += __shfl_xor(s2, 16, 32); s3 += __shfl_xor(s3, 16, 32);
    const int n = n0 + ll;
    if (hi == 0 && n < n1b) {
      float e;
      float s;
      s = s0 + csh[0]; e = (s > 0.f) ? s : SLOPE * s; e4[(size_t)n * H + 0] = e; mh0 = fmaxf(mh0, e);
      s = s1 + csh[1]; e = (s > 0.f) ? s : SLOPE * s; e4[(size_t)n * H + 1] = e; mh1 = fmaxf(mh1, e);
      s = s2 + csh[2]; e = (s > 0.f) ? s : SLOPE * s; e4[(size_t)n * H + 2] = e; mh2 = fmaxf(mh2, e);
      s = s3 + csh[3]; e = (s > 0.f) ? s : SLOPE * s; e4[(size_t)n * H + 3] = e; mh3 = fmaxf(mh3, e);
    }
  }
#pragma unroll
  for (int off = 16; off; off >>= 1) {
    mh0 = fmaxf(mh0, __shfl_xor(mh0, off, 32));
    mh1 = fmaxf(mh1, __shfl_xor(mh1, off, 32));
    mh2 = fmaxf(mh2, __shfl_xor(mh2, off, 32));
    mh3 = fmaxf(mh3, __shfl_xor(mh3, off, 32));
  }
  if (lane == 0) {
    wmaxsh[wave * H + 0] = mh0; wmaxsh[wave * H + 1] = mh1;
    wmaxsh[wave * H + 2] = mh2; wmaxsh[wave * H + 3] = mh3;
  }
#endif
  __syncthreads();
  if (tid < H) {
    float m = -INFINITY;
    for (int w = 0; w < 8; ++w) m = fmaxf(m, wmaxsh[w * H + tid]);
    blkmax[blockIdx.x * H + tid] = m;
  }
}

// Kernel 2: global per-head max over block partials (tiny).
__global__ void maxred_kernel(const float* __restrict__ blkmax,
                              float* __restrict__ mg) {
  __shared__ float red[256];
  const int tid = threadIdx.x;
  const int h = tid & 3;
  float m = -INFINITY;
  for (int b = tid >> 2; b < B1; b += 64) m = fmaxf(m, blkmax[b * H + h]);
  red[tid] = m;
  __syncthreads();
  if (tid < H) {
    float mm = -INFINITY;
    for (int j = 0; j < 64; ++j) mm = fmaxf(mm, red[j * H + tid]);
    mg[tid] = mm;
  }
}

// ---------------------------------------------------------------------------
// Kernel 3 (streaming pass 2): per block, stage p = exp(e - m) in LDS, then
// walk one full 2KB code row per iteration (256 threads x b64, fully
// coalesced); each thread owns 2 columns x 4 heads in registers. No atomics.
// ---------------------------------------------------------------------------
__global__ void weight_kernel(const float* __restrict__ code,
                              const float* __restrict__ e4,
                              const float* __restrict__ mg,
                              float* __restrict__ dpart,
                              float* __restrict__ wpart) {
  __shared__ float msh[H];
  __shared__ float p_lds[CHUNK * H];
  const int tid = threadIdx.x;
  const int n0 = blockIdx.x * CHUNK;
  const int n1 = min(n0 + CHUNK, NCODES);
  const int cnt = (n1 > n0) ? (n1 - n0) : 0;
  if (tid < H) msh[tid] = mg[tid];
  __syncthreads();
  for (int idx = tid; idx < cnt * H; idx += 256)
    p_lds[idx] = __expf(e4[(size_t)n0 * H + idx] - msh[idx & 3]);
  __syncthreads();
  if (tid < H) {
    float s = 0.f;
    for (int i = 0; i < cnt; ++i) s += p_lds[i * H + tid];
    dpart[blockIdx.x * H + tid] = s;
  }
  float a00 = 0, a01 = 0, a10 = 0, a11 = 0, a20 = 0, a21 = 0, a30 = 0, a31 = 0;
  const float* cp = code + (size_t)n0 * D + tid * 2;
  for (int i = 0; i < cnt; ++i) {
    v2f cv = *(const v2f*)(cp + (size_t)i * D);
    const float p0 = p_lds[i * H + 0], p1 = p_lds[i * H + 1];
    const float p2 = p_lds[i * H + 2], p3 = p_lds[i * H + 3];
    a00 += p0 * cv.x; a01 += p0 * cv.y;
    a10 += p1 * cv.x; a11 += p1 * cv.y;
    a20 += p2 * cv.x; a21 += p2 * cv.y;
    a30 += p3 * cv.x; a31 += p3 * cv.y;
  }
  float* wp = wpart + (size_t)blockIdx.x * (H * D);
  wp[0 * D + tid * 2] = a00; wp[0 * D + tid * 2 + 1] = a01;
  wp[1 * D + tid * 2] = a10; wp[1 * D + tid * 2 + 1] = a11;
  wp[2 * D + tid * 2] = a20; wp[2 * D + tid * 2 + 1] = a21;
  wp[3 * D + tid * 2] = a30; wp[3 * D + tid * 2 + 1] = a31;
}

// Kernel 4: reduce per-block weighted sums (fixed order -> deterministic).
__global__ void wsum_kernel(const float* __restrict__ wpart,
                            float* __restrict__ wsum) {
  const int hd = blockIdx.x * 256 + threadIdx.x;
  float s = 0.f;
  for (int b = 0; b < B1; ++b) s += wpart[(size_t)b * (H * D) + hd];
  wsum[hd] = s;
}

// Kernel 5: attended[h] = (wsum[h]/denom[h]) @ W[h]; writes mho to d_out+1024.
// 8 blocks (h, k-half); coalesced column walk of W.
__global__ void attend_kernel(const float* __restrict__ W,
                              const float* __restrict__ wsum,
                              const float* __restrict__ dpart,
                              float* __restrict__ outp) {
  __shared__ float winv[D];
  __shared__ float dred[256];
  __shared__ float dsh;
  const int tid = threadIdx.x;
  const int h = blockIdx.x >> 1;
  const int k0 = (blockIdx.x & 1) * 256;
  float part = 0.f;
  for (int b = tid; b < B1; b += 256) part += dpart[b * H + h];
  dred[tid] = part;
  __syncthreads();
  if (tid == 0) {
    float s = 0.f;
    for (int j = 0; j < 256; ++j) s += dred[j];
    dsh = s;
  }
  __syncthreads();
  const float inv = 1.0f / dsh;
  winv[tid]       = wsum[h * D + tid] * inv;
  winv[tid + 256] = wsum[h * D + tid + 256] * inv;
  __syncthreads();
  float acc = 0.f;
  const float* wb = W + (size_t)h * D * D + k0 + tid;
  for (int d = 0; d < D; ++d) acc += winv[d] * wb[(size_t)d * D];
  outp[NSPEC + h * D + k0 + tid] = acc;
}

// Kernel 6: logits partials over (j-slice x s-slice) -> 256 blocks for BW.
__global__ void logits_part_kernel(const float* __restrict__ Wc,
                                   const float* __restrict__ mho,
                                   float* __restrict__ lpart) {
  __shared__ float red[256];
  const int tid = threadIdx.x;
  const int jg = blockIdx.x & 7;
  const int sb = blockIdx.x >> 3;
  const int sl = tid & 31, jsub = tid >> 5;
  const int s = sb * 32 + sl;
  const int jbase = jg * 256 + jsub * 32;
  float acc = 0.f;
  for (int jj = 0; jj < 32; ++jj) {
    const int j = jbase + jj;
    acc += mho[j] * Wc[(size_t)j * NSPEC + s];
  }
  red[tid] = acc;
  __syncthreads();
  if (tid < 32) {
    float tot = 0.f;
    for (int q = 0; q < 8; ++q) tot += red[q * 32 + tid];
    lpart[jg * NSPEC + sb * 32 + tid] = tot;
  }
}

// Kernel 7: logits[s] = bc[s] + sum of 8 partials.
__global__ void logits_final_kernel(const float* __restrict__ lpart,
                                    const float* __restrict__ bc,
                                    float* __restrict__ outp) {
  const int s = blockIdx.x * 256 + threadIdx.x;
  float acc = bc[s];
  for (int jg = 0; jg < 8; ++jg) acc += lpart[jg * NSPEC + s];
  outp[s] = acc;
}

extern "C" void kernel_launch(void* const* d_in, const int* in_sizes, int n_in,
                              void* d_out, int out_size, void* d_ws, size_t ws_size,
                              hipStream_t stream) {
  (void)in_sizes; (void)n_in; (void)out_size; (void)ws_size;
  const float* prov = (const float*)d_in[0];   // [1,512]
  const float* code = (const float*)d_in[1];   // [100000,512]
  const float* W    = (const float*)d_in[2];   // [4,512,512]
  const float* a1   = (const float*)d_in[3];   // [4,512]
  const float* a2   = (const float*)d_in[4];   // [4,512]
  const float* Wc   = (const float*)d_in[5];   // [2048,1024]
  const float* bc   = (const float*)d_in[6];   // [1024]
  float* out = (float*)d_out;                  // [0,1024)=logits, [1024,3072)=mho

  // workspace layout (~5.9 MB of floats)
  float* w      = (float*)d_ws;
  float* v2g    = w;                           // H*D
  float* tt     = v2g + H * D;                 // H*D
  float* e4     = tt + H * D;                  // NCODES*H
  float* blkmax = e4 + (size_t)NCODES * H;     // B1*H
  float* mg     = blkmax + B1 * H;             // H
  float* dpart  = mg + H;                      // B1*H
  float* wpart  = dpart + B1 * H;              // B1*H*D
  float* wsum   = wpart + (size_t)B1 * H * D;  // H*D
  float* lpart  = wsum + H * D;                // 8*NSPEC

  prep_kernel<<<64, 256, 0, stream>>>(prov, W, a1, a2, v2g, tt);
  score_kernel<<<B1, 256, 0, stream>>>(code, v2g, tt, e4, blkmax);
  maxred_kernel<<<1, 256, 0, stream>>>(blkmax, mg);
  weight_kernel<<<B1, 256, 0, stream>>>(code, e4, mg, dpart, wpart);
  wsum_kernel<<<8, 256, 0, stream>>>(wpart, wsum);
  attend_kernel<<<8, 256, 0, stream>>>(W, wsum, dpart, out);
  logits_part_kernel<<<256, 256, 0, stream>>>(Wc, out + NSPEC, lpart);
  logits_final_kernel<<<4, 256, 0, stream>>>(lpart, bc, out);
}